// GOSR_38912403702236
// MI455X (gfx1250) — compile-verified
//
#include <hip/hip_runtime.h>
#include <math.h>

#define D   64
#define DEG 50
#define NU  32768
#define NI  16384

typedef __attribute__((ext_vector_type(2))) float v2f;
typedef __attribute__((ext_vector_type(4))) float v4f;
typedef __attribute__((ext_vector_type(8))) float v8f;
typedef __attribute__((ext_vector_type(4))) int   v4i;

// pointer-to-v4i in global (AS1) and LDS (AS3) address spaces
typedef __attribute__((address_space(1))) v4i* gv4i_p;
typedef __attribute__((address_space(3))) v4i* lv4i_p;

// ---------------- wave32 reductions (gfx1250 is wave32-only) ----------------
__device__ __forceinline__ float wave_sum(float v) {
#pragma unroll
  for (int m = 16; m > 0; m >>= 1) v += __shfl_xor(v, m, 32);
  return v;
}
__device__ __forceinline__ float wave_max(float v) {
#pragma unroll
  for (int m = 16; m > 0; m >>= 1) v = fmaxf(v, __shfl_xor(v, m, 32));
  return v;
}

// -------- gfx1250 async global->LDS gather (probe via __has_builtin) --------
#if defined(__has_builtin)
#if __has_builtin(__builtin_amdgcn_global_load_async_to_lds_b128)
#define ATTN_ASYNC_GATHER 1
#endif
#endif

__device__ __forceinline__ void wait_asynccnt0() {
#if defined(__has_builtin)
#if __has_builtin(__builtin_amdgcn_s_wait_asynccnt)
  __builtin_amdgcn_s_wait_asynccnt(0);
  return;
#endif
#endif
#if defined(__gfx1250__)
  asm volatile("s_wait_asynccnt 0x0" ::: "memory");
#endif
}

#define WMMA_F32(a, b, c) \
  __builtin_amdgcn_wmma_f32_16x16x4_f32(false, (a), false, (b), (short)0, (c), false, false)

// ---------------------------------------------------------------------------
// Out[rows,64] = X[rows,kdim] @ W[64,kdim]^T   via V_WMMA_F32_16X16X4_F32.
// One wave per 16-row strip; 4 accumulators (all 4 col tiles) share each A
// fragment. Each lane loads b128 (4 floats at k8 + 4*hi) which feeds TWO
// WMMA k-steps under a permuted-K mapping:
//   instance even: slots (x,y) -> physical K {k8,k8+1} (low) u {k8+4,k8+5} (high)
//   instance odd : slots (z,w) -> physical K {k8+2,k8+3} u {k8+6,k8+7}
// A and B use the identical mapping, so the reduction covers each K once.
// Fragment layout assumptions (ISA 7.12.2):
//   A (16x4): lanes 0-15 M=lane {v0=K0,v1=K1}; lanes 16-31 {v0=K2,v1=K3}.
//   B (4x16): mirrored half-wave K split.
//   C/D:      VGPR r, lanes 0-15 -> (M=r, N=lane); lanes 16-31 -> (M=r+8).
// Optional epilogue: out = elu(acc + resid)  (alpha = 1).
// ---------------------------------------------------------------------------
__global__ void __launch_bounds__(128)
gemm_wmma(const float* __restrict__ X, const float* __restrict__ W,
          const float* __restrict__ resid, float* __restrict__ out,
          int kdim, int apply_elu) {
  const int lane = threadIdx.x & 31;
  const int wave = threadIdx.x >> 5;
  const int m0   = (blockIdx.x * 4 + wave) * 16;   // 16-row strip per wave
  const int l    = lane & 15;
  const int hi   = (lane >= 16);
  const int koff = hi ? 4 : 0;                     // b128 per lane-half

  const float* xr  = X + (size_t)(m0 + l) * kdim + koff;
  const float* wr0 = W + (size_t)(0 * 16 + l) * kdim + koff;
  const float* wr1 = W + (size_t)(1 * 16 + l) * kdim + koff;
  const float* wr2 = W + (size_t)(2 * 16 + l) * kdim + koff;
  const float* wr3 = W + (size_t)(3 * 16 + l) * kdim + koff;

  v8f acc0 = {}, acc1 = {}, acc2 = {}, acc3 = {};
  for (int k8 = 0; k8 < kdim; k8 += 8) {
    const v4f A  = *(const v4f*)(xr  + k8);
    const v4f B0 = *(const v4f*)(wr0 + k8);
    const v4f B1 = *(const v4f*)(wr1 + k8);
    const v4f B2 = *(const v4f*)(wr2 + k8);
    const v4f B3 = *(const v4f*)(wr3 + k8);
    v2f a0; a0.x = A.x; a0.y = A.y;
    v2f a1; a1.x = A.z; a1.y = A.w;
    v2f b;
    b.x = B0.x; b.y = B0.y; acc0 = WMMA_F32(a0, b, acc0);
    b.x = B1.x; b.y = B1.y; acc1 = WMMA_F32(a0, b, acc1);
    b.x = B2.x; b.y = B2.y; acc2 = WMMA_F32(a0, b, acc2);
    b.x = B3.x; b.y = B3.y; acc3 = WMMA_F32(a0, b, acc3);
    b.x = B0.z; b.y = B0.w; acc0 = WMMA_F32(a1, b, acc0);
    b.x = B1.z; b.y = B1.w; acc1 = WMMA_F32(a1, b, acc1);
    b.x = B2.z; b.y = B2.w; acc2 = WMMA_F32(a1, b, acc2);
    b.x = B3.z; b.y = B3.w; acc3 = WMMA_F32(a1, b, acc3);
  }

  const int mbase = m0 + (hi ? 8 : 0);
  v8f acc[4] = {acc0, acc1, acc2, acc3};
#pragma unroll
  for (int nt = 0; nt < 4; ++nt) {
    const int col = nt * 16 + l;
#pragma unroll
    for (int r = 0; r < 8; ++r) {
      const size_t idx = (size_t)(mbase + r) * D + col;
      float v = acc[nt][r];
      if (apply_elu) {
        v += resid[idx];
        v = (v > 0.0f) ? v : (expf(v) - 1.0f);
      }
      out[idx] = v;
    }
  }
}

// ---------------------------------------------------------------------------
// Per-node 'orgat' + 'att' aggregation. One wave32 per node; each lane owns
// feature dims (lane, lane+32). Mailbox (50x64 f32 = 12.8KB) staged in LDS
// via async global->LDS B128 (2 rows per instruction: lanes 0-15 -> row k,
// lanes 16-31 -> row k+1; 16B per lane), tracked by ASYNCcnt.
// Writes the gated concat input  cat[node] = [h_long(64) | h_short(64)].
// ---------------------------------------------------------------------------
__global__ void __launch_bounds__(32)
attn_kernel(const float* __restrict__ Hsrc,   // [Nsrc,64] projected source table
            const float* __restrict__ Hdst,   // [N,64]    projected dst embeddings
            const int*   __restrict__ nbr,    // [N,DEG]
            const int*   __restrict__ times,  // [N,DEG]
            const float* __restrict__ te,     // [DEG,64]
            const float* __restrict__ tek,    // [DEG,64]
            float* __restrict__ cat) {        // [N,128]
  __shared__ float mail[DEG][D];
  __shared__ int   ts[DEG];
  __shared__ int   ro[DEG];
  __shared__ float e0[DEG];
  __shared__ float e1[DEG];
  __shared__ int   last_s;

  const int node = blockIdx.x;
  const int lane = threadIdx.x;

  for (int k = lane; k < DEG; k += 32) ts[k] = times[(size_t)node * DEG + k];
  __syncthreads();

  // re_order[k] = DEG - 1 - rank(times[k]); stable-sort tie-break (j < k).
  for (int k = lane; k < DEG; k += 32) {
    const int tk = ts[k];
    int rank = 0;
    for (int j = 0; j < DEG; ++j) {
      const int tj = ts[j];
      rank += (tj < tk) || (tj == tk && j < k);
    }
    ro[k] = DEG - 1 - rank;
  }
  if (lane == 0) {                 // argmax (first occurrence, like jnp.argmax)
    int best = ts[0], bi = 0;
    for (int j = 1; j < DEG; ++j)
      if (ts[j] > best) { best = ts[j]; bi = j; }
    last_s = bi;
  }

  // ---- gather mailbox rows from the L2-resident projected table into LDS ----
#ifdef ATTN_ASYNC_GATHER
  {
    const int r_off = lane >> 4;              // lanes 0-15: row k; 16-31: row k+1
    const int col   = (lane & 15) * 4;        // 4 floats (16B) per lane
    for (int k = 0; k < DEG; k += 2) {
      const int r   = k + r_off;
      const int src = nbr[(size_t)node * DEG + r];
      const float* g = Hsrc + (size_t)src * D + col;
      __builtin_amdgcn_global_load_async_to_lds_b128(
          (gv4i_p)g, (lv4i_p)&mail[r][col], /*offset=*/0, /*cpol=*/0);
    }
    wait_asynccnt0();
  }
#else
  for (int k = 0; k < DEG; ++k) {
    const int src = nbr[(size_t)node * DEG + k];
    const float* row = Hsrc + (size_t)src * D;
    mail[k][lane]      = row[lane];
    mail[k][lane + 32] = row[lane + 32];
  }
#endif
  __syncthreads();

  const float d1 = Hdst[(size_t)node * D + lane];
  const float d2 = Hdst[(size_t)node * D + lane + 32];
  const int   last = last_s;
  const float le1 = mail[last][lane];
  const float le2 = mail[last][lane + 32];
  const float inv_scale = 0.125f;          // 1/sqrt(64)

  for (int k = 0; k < DEG; ++k) {
    const int   r  = ro[k];
    const float m1 = mail[k][lane];
    const float m2 = mail[k][lane + 32];
    float p = (te[r * D + lane] + m1) * d1 + (te[r * D + lane + 32] + m2) * d2;
    float q = m1 * le1 + m2 * le2;
    p = wave_sum(p);
    q = wave_sum(q);
    if (lane == 0) { e0[k] = p * inv_scale; e1[k] = q * inv_scale; }
  }
  __syncthreads();

  // two softmaxes over DEG
  float mx0 = -__builtin_inff(), mx1 = -__builtin_inff();
  for (int k = lane; k < DEG; k += 32) {
    mx0 = fmaxf(mx0, e0[k]);
    mx1 = fmaxf(mx1, e1[k]);
  }
  mx0 = wave_max(mx0);
  mx1 = wave_max(mx1);
  float s0 = 0.0f, s1 = 0.0f;
  for (int k = lane; k < DEG; k += 32) {
    const float a = expf(e0[k] - mx0); e0[k] = a; s0 += a;
    const float b = expf(e1[k] - mx1); e1[k] = b; s1 += b;
  }
  s0 = wave_sum(s0);
  s1 = wave_sum(s1);
  const float r0 = 1.0f / s0, r1 = 1.0f / s1;
  __syncthreads();

  float hl1 = 0.0f, hl2 = 0.0f, hs1 = 0.0f, hs2 = 0.0f;
  for (int k = 0; k < DEG; ++k) {
    const int   r  = ro[k];
    const float a  = e0[k] * r0;
    const float b  = e1[k] * r1;
    const float m1 = mail[k][lane];
    const float m2 = mail[k][lane + 32];
    hl1 += a * (m1 + tek[r * D + lane]);
    hl2 += a * (m2 + tek[r * D + lane + 32]);
    hs1 += b * m1;
    hs2 += b * m2;
  }
  float* c = cat + (size_t)node * (2 * D);
  c[lane]      = hl1;
  c[lane + 32] = hl2;
  c[64 + lane] = hs1;
  c[96 + lane] = hs2;
}

// ---------------------------------------------------------------------------
extern "C" void kernel_launch(void* const* d_in, const int* in_sizes, int n_in,
                              void* d_out, int out_size, void* d_ws, size_t ws_size,
                              hipStream_t stream) {
  (void)in_sizes; (void)n_in; (void)out_size; (void)ws_size;

  const float* user_feat = (const float*)d_in[0];
  const float* item_feat = (const float*)d_in[1];
  const float* W_user    = (const float*)d_in[2];
  const float* W_item    = (const float*)d_in[3];
  const float* gate_u    = (const float*)d_in[4];
  const float* gate_i    = (const float*)d_in[5];
  const float* u_te      = (const float*)d_in[6];
  const float* u_tek     = (const float*)d_in[7];
  const float* i_te      = (const float*)d_in[8];
  const float* i_tek     = (const float*)d_in[9];
  const int*   user_nbr  = (const int*)d_in[10];
  const int*   item_nbr  = (const int*)d_in[11];
  const int*   user_time = (const int*)d_in[12];
  const int*   item_time = (const int*)d_in[13];

  float* ws       = (float*)d_ws;
  float* user_h   = ws;                                  // [NU,64]
  float* item_h   = user_h + (size_t)NU * D;             // [NI,64]
  float* user_cat = item_h + (size_t)NI * D;             // [NU,128]
  float* item_cat = user_cat + (size_t)NU * 2 * D;       // [NI,128]

  float* user_out = (float*)d_out;                       // [NU,64]
  float* item_out = user_out + (size_t)NU * D;           // [NI,64]

  // 1) fp32 WMMA projections into workspace (tables then live in L2)
  gemm_wmma<<<NU / 64, 128, 0, stream>>>(user_feat, W_user, nullptr, user_h, D, 0);
  gemm_wmma<<<NI / 64, 128, 0, stream>>>(item_feat, W_item, nullptr, item_h, D, 0);

  // 2) per-node attention aggregation (gather-bound; mailbox staged in LDS)
  attn_kernel<<<NI, 32, 0, stream>>>(user_h, item_h, item_nbr, item_time,
                                     i_te, i_tek, item_cat);
  attn_kernel<<<NU, 32, 0, stream>>>(item_h, user_h, user_nbr, user_time,
                                     u_te, u_tek, user_cat);

  // 3) fp32 WMMA gate matmul with fused residual + ELU epilogue -> d_out
  gemm_wmma<<<NU / 64, 128, 0, stream>>>(user_cat, gate_u, user_feat, user_out, 2 * D, 1);
  gemm_wmma<<<NI / 64, 128, 0, stream>>>(item_cat, gate_i, item_feat, item_out, 2 * D, 1);
}